// MambaBlock_12481174962850
// MI455X (gfx1250) — compile-verified
//
#include <hip/hip_runtime.h>
#include <hip/hip_bf16.h>
#include <math.h>

#define B_SZ 4
#define SEQ 1024
#define D_MODEL 1024
#define D_STATE 16
#define D_CONV 4
#define D_INNER 2048            // EXPAND * D_MODEL
#define ROWS (B_SZ * SEQ)       // 4096

typedef float v2f __attribute__((ext_vector_type(2)));
typedef float v8f __attribute__((ext_vector_type(8)));

__device__ __forceinline__ float sigmoidf_(float x) { return 1.f / (1.f + __expf(-x)); }

// ---------------------------------------------------------------------------
// LayerNorm (torch-style: alpha*(x-mean)/(std+eps)+bias, std with ddof=1).
// Optional residual addend (nullptr for the input norm). One block per row.
// ---------------------------------------------------------------------------
__global__ __launch_bounds__(256) void ln_kernel(
    const float* __restrict__ x, const float* __restrict__ addend,
    const float* __restrict__ alpha, const float* __restrict__ beta,
    float* __restrict__ out) {
  const int D = D_MODEL;
  const int row = blockIdx.x;
  const int tid = threadIdx.x;
  const float* xr = x + (size_t)row * D;
  const float* ar = addend ? addend + (size_t)row * D : nullptr;
  float v[4];
  float s = 0.f, s2 = 0.f;
#pragma unroll
  for (int i = 0; i < 4; ++i) {
    int d = tid + i * 256;
    float t = xr[d];
    if (ar) t += ar[d];
    v[i] = t;
    s += t;
    s2 += t * t;
  }
  __shared__ float r1[256], r2[256];
  r1[tid] = s;
  r2[tid] = s2;
  __syncthreads();
  for (int off = 128; off > 0; off >>= 1) {
    if (tid < off) {
      r1[tid] += r1[tid + off];
      r2[tid] += r2[tid + off];
    }
    __syncthreads();
  }
  float mean = r1[0] / (float)D;
  float var = (r2[0] - (float)D * mean * mean) / (float)(D - 1);
  float inv = 1.f / (sqrtf(fmaxf(var, 0.f)) + 1e-6f);
#pragma unroll
  for (int i = 0; i < 4; ++i) {
    int d = tid + i * 256;
    out[(size_t)row * D + d] = alpha[d] * (v[i] - mean) * inv + beta[d];
  }
}

// ---------------------------------------------------------------------------
// Generic fp32 WMMA GEMM:  C[M,N] = A[M,K] @ W[N,K]^T + bias[N]
// V_WMMA_F32_16X16X4_F32; wave tile = 16 x (16*JT); 8 waves/block = 128 rows.
// JT is compile-time so every fragment load is unconditional (EXEC all-ones,
// no scalar branches in the hot loop). K loop is software-pipelined in steps
// of 8 (two WMMA groups per iteration, next step's fragments loaded first).
// Requires: M % 128 == 0, N % (16*JT) == 0, K % 8 == 0.
// EPI: 0 = bias only, 1 = bias + softplus + clip[1e-4, 10] (delta path).
//
// Fragment layouts per CDNA5 ISA (32-bit 16x4 A / 4x16 B / 16x16 C):
//   lane = lo + 16*hi;  A vgpr j holds A[lo][k + 2*hi + j]
//                       B vgpr j holds B[k + 2*hi + j][col=lo]  (= W[col][k..])
//                       C vgpr r holds C[m0 + r + 8*hi][lo]
// ---------------------------------------------------------------------------
template <int JT, int EPI>
__global__ __launch_bounds__(256) void gemm_wmma_kernel(
    const float* __restrict__ A, const float* __restrict__ W,
    const float* __restrict__ bias, float* __restrict__ C,
    int M, int N, int K) {
  const int lane = threadIdx.x & 31;
  const int wave = threadIdx.x >> 5;
  const int lo = lane & 15;
  const int hi = lane >> 4;
  const int m0 = blockIdx.x * 128 + wave * 16;
  const int n0 = blockIdx.y * (16 * JT);

  v8f acc[JT];
#pragma unroll
  for (int j = 0; j < JT; ++j) acc[j] = (v8f){0.f, 0.f, 0.f, 0.f, 0.f, 0.f, 0.f, 0.f};

  const float* arow = A + (size_t)(m0 + lo) * K + 2 * hi;
  const float* wp[JT];
#pragma unroll
  for (int j = 0; j < JT; ++j) wp[j] = W + (size_t)(n0 + j * 16 + lo) * K + 2 * hi;

  // ---- software pipeline: stage 0 ----
  v2f a[2];
  v2f b[2][JT];
  a[0] = *(const v2f*)(arow + 0);
  a[1] = *(const v2f*)(arow + 4);
#pragma unroll
  for (int j = 0; j < JT; ++j) {
    b[0][j] = *(const v2f*)(wp[j] + 0);
    b[1][j] = *(const v2f*)(wp[j] + 4);
  }

  for (int k = 0; k + 8 < K; k += 8) {
    // load next step's fragments (in flight while WMMAs execute)
    v2f an[2];
    v2f bn[2][JT];
    an[0] = *(const v2f*)(arow + k + 8);
    an[1] = *(const v2f*)(arow + k + 12);
#pragma unroll
    for (int j = 0; j < JT; ++j) {
      bn[0][j] = *(const v2f*)(wp[j] + k + 8);
      bn[1][j] = *(const v2f*)(wp[j] + k + 12);
    }
    __builtin_prefetch(arow + k + 64, 0, 0);
#pragma unroll
    for (int u = 0; u < 2; ++u)
#pragma unroll
      for (int j = 0; j < JT; ++j)
        acc[j] = __builtin_amdgcn_wmma_f32_16x16x4_f32(
            false, a[u], false, b[u][j], (short)0, acc[j], false, false);
    a[0] = an[0];
    a[1] = an[1];
#pragma unroll
    for (int j = 0; j < JT; ++j) {
      b[0][j] = bn[0][j];
      b[1][j] = bn[1][j];
    }
  }
  // ---- pipeline drain: last K-step ----
#pragma unroll
  for (int u = 0; u < 2; ++u)
#pragma unroll
    for (int j = 0; j < JT; ++j)
      acc[j] = __builtin_amdgcn_wmma_f32_16x16x4_f32(
          false, a[u], false, b[u][j], (short)0, acc[j], false, false);

  // ---- epilogue ----
#pragma unroll
  for (int j = 0; j < JT; ++j) {
    int col = n0 + j * 16 + lo;
    float bv = bias[col];
#pragma unroll
    for (int r = 0; r < 8; ++r) {
      int m = m0 + 8 * hi + r;
      float vv = acc[j][r] + bv;
      if (EPI == 1) {
        vv = (vv > 20.f) ? vv : log1pf(__expf(vv));  // softplus
        vv = fminf(fmaxf(vv, 1e-4f), 10.f);          // clip
      }
      C[(size_t)m * N + col] = vv;
    }
  }
}

// ---------------------------------------------------------------------------
// Causal depthwise conv (d_conv=4) + SiLU. x_in is the first D_INNER columns
// of xz (row stride 2*D_INNER). One thread per (b,t,e), e fastest (coalesced).
// ---------------------------------------------------------------------------
__global__ __launch_bounds__(256) void conv_silu_kernel(
    const float* __restrict__ xz, const float* __restrict__ cw,
    const float* __restrict__ cb, float* __restrict__ xconv) {
  int idx = blockIdx.x * blockDim.x + threadIdx.x;   // < ROWS*D_INNER
  int e = idx & (D_INNER - 1);
  int bt = idx >> 11;                                 // D_INNER == 2048
  int t = bt & (SEQ - 1);
  float acc = cb[e];
#pragma unroll
  for (int j = 0; j < D_CONV; ++j) {
    int tt = t - (D_CONV - 1) + j;
    if (tt >= 0)
      acc += cw[e * D_CONV + j] *
             xz[(size_t)(bt - (D_CONV - 1) + j) * (2 * D_INNER) + e];
  }
  xconv[idx] = acc * sigmoidf_(acc);
}

// ---------------------------------------------------------------------------
// Selective scan, fused with SiLU(z) gating. One thread per (b,e); h[16] and
// -exp(A_log[e,:]) live in registers. B/C rows for 16 timesteps are staged in
// LDS per chunk (broadcast reads, conflict-free). Renorm every 16 steps is
// applied BEFORE the y dot, matching the reference step order.
// ---------------------------------------------------------------------------
__global__ __launch_bounds__(256) void scan_kernel(
    const float* __restrict__ xconv, const float* __restrict__ delta,
    const float* __restrict__ bc, const float* __restrict__ A_log,
    const float* __restrict__ Dp, const float* __restrict__ xz,
    float* __restrict__ y) {
  const int E = D_INNER, N = D_STATE, L = SEQ;
  const int b = blockIdx.x >> 3;                 // E/256 == 8 chunks
  const int e = ((blockIdx.x & 7) << 8) + threadIdx.x;

  float negA[D_STATE];
#pragma unroll
  for (int n = 0; n < N; ++n) negA[n] = -__expf(A_log[e * N + n]);
  float Dv = Dp[e];
  float h[D_STATE];
#pragma unroll
  for (int n = 0; n < N; ++n) h[n] = 0.f;

  __shared__ float sB[16][D_STATE];
  __shared__ float sC[16][D_STATE];

  for (int tc = 0; tc < L / 16; ++tc) {
    int base = (b * L + tc * 16) * (2 * N);
    for (int i = threadIdx.x; i < 16 * 2 * N; i += blockDim.x) {
      int row = i >> 5, col = i & 31;
      float vv = bc[base + i];
      if (col < N) sB[row][col] = vv;
      else         sC[row][col - N] = vv;
    }
    __syncthreads();

#pragma unroll 1
    for (int s = 0; s < 16; ++s) {
      int t = tc * 16 + s;
      int idx = (b * L + t) * E + e;
      float xt = xconv[idx];
      float dt = delta[idx];
#pragma unroll
      for (int n = 0; n < N; ++n) {
        float da = __expf(fminf(fmaxf(dt * negA[n], -10.f), 0.f));
        h[n] = da * h[n] + (dt * sB[s][n]) * xt;
      }
      if (s == 15) {  // periodic renormalization (t % 16 == 15)
        float ss = 0.f;
#pragma unroll
        for (int n = 0; n < N; ++n) ss += h[n] * h[n];
        float hn = fmaxf(sqrtf(ss), 1e-6f);
        float scale = (hn > 10.f) ? (10.f / hn) : 1.f;
#pragma unroll
        for (int n = 0; n < N; ++n) h[n] *= scale;
      }
      float yv = Dv * xt;
#pragma unroll
      for (int n = 0; n < N; ++n) yv += sC[s][n] * h[n];
      float z = xz[(size_t)(b * L + t) * (2 * E) + E + e];
      y[idx] = yv * (z * sigmoidf_(z));
    }
    __syncthreads();
  }
}

// ---------------------------------------------------------------------------
// Host launcher
// ---------------------------------------------------------------------------
extern "C" void kernel_launch(void* const* d_in, const int* in_sizes, int n_in,
                              void* d_out, int out_size, void* d_ws, size_t ws_size,
                              hipStream_t stream) {
  const float* x            = (const float*)d_in[0];
  const float* in_proj_w    = (const float*)d_in[1];
  const float* in_proj_b    = (const float*)d_in[2];
  const float* conv_w       = (const float*)d_in[3];
  const float* conv_b       = (const float*)d_in[4];
  const float* x_proj_w     = (const float*)d_in[5];
  const float* x_proj_b     = (const float*)d_in[6];
  const float* dt_proj_w    = (const float*)d_in[7];
  const float* dt_proj_b    = (const float*)d_in[8];
  const float* A_log        = (const float*)d_in[9];
  const float* D_param      = (const float*)d_in[10];
  const float* out_proj_w   = (const float*)d_in[11];
  const float* out_proj_b   = (const float*)d_in[12];
  const float* norm_alpha   = (const float*)d_in[13];
  const float* norm_bias    = (const float*)d_in[14];
  const float* in_norm_alpha = (const float*)d_in[15];
  const float* in_norm_bias  = (const float*)d_in[16];
  float* out = (float*)d_out;

  float* ws = (float*)d_ws;
  float* xn    = ws;                               // ROWS*D_MODEL  (reused as out_proj result)
  float* xz    = xn    + (size_t)ROWS * D_MODEL;   // ROWS*2E
  float* xconv = xz    + (size_t)ROWS * 2 * D_INNER;
  float* bcbuf = xconv + (size_t)ROWS * D_INNER;   // ROWS*2N
  float* dlt   = bcbuf + (size_t)ROWS * 2 * D_STATE;
  float* ybuf  = dlt   + (size_t)ROWS * D_INNER;

  // 1) input layernorm
  ln_kernel<<<ROWS, 256, 0, stream>>>(x, nullptr, in_norm_alpha, in_norm_bias, xn);

  // 2) in_proj: xz = xn @ in_proj_w^T + b        (4096 x 4096 x 1024)
  gemm_wmma_kernel<4, 0><<<dim3(ROWS / 128, (2 * D_INNER) / 64), 256, 0, stream>>>(
      xn, in_proj_w, in_proj_b, xz, ROWS, 2 * D_INNER, D_MODEL);

  // 3) causal depthwise conv + SiLU
  conv_silu_kernel<<<(ROWS * D_INNER) / 256, 256, 0, stream>>>(xz, conv_w, conv_b, xconv);

  // 4) x_proj: bc = xconv @ x_proj_w^T + b       (4096 x 32 x 2048), JT=2 tile
  gemm_wmma_kernel<2, 0><<<dim3(ROWS / 128, 1), 256, 0, stream>>>(
      xconv, x_proj_w, x_proj_b, bcbuf, ROWS, 2 * D_STATE, D_INNER);

  // 5) dt_proj + softplus + clip: delta          (4096 x 2048 x 2048)
  gemm_wmma_kernel<4, 1><<<dim3(ROWS / 128, D_INNER / 64), 256, 0, stream>>>(
      xconv, dt_proj_w, dt_proj_b, dlt, ROWS, D_INNER, D_INNER);

  // 6) selective scan + SiLU(z) gating
  scan_kernel<<<B_SZ * (D_INNER / 256), 256, 0, stream>>>(
      xconv, dlt, bcbuf, A_log, D_param, xz, ybuf);

  // 7) out_proj: xn(reused) = ybuf @ out_proj_w^T + b   (4096 x 1024 x 2048)
  gemm_wmma_kernel<4, 0><<<dim3(ROWS / 128, D_MODEL / 64), 256, 0, stream>>>(
      ybuf, out_proj_w, out_proj_b, xn, ROWS, D_MODEL, D_INNER);

  // 8) residual add + final layernorm
  ln_kernel<<<ROWS, 256, 0, stream>>>(x, xn, norm_alpha, norm_bias, out);
}